// FeedForwardQuantum_65481071403168
// MI455X (gfx1250) — compile-verified
//
#include <hip/hip_runtime.h>
#include <hip/hip_bf16.h>

// ---------------------------------------------------------------------------
// Fused "quantum FFN":  out = relu( (cos(x[...,:8]) * cos(theta)) @ W1^T ) @ W2^T
// fp32 end-to-end via V_WMMA_F32_16X16X4_F32.
//
// Tiling rationale (MI455X):
//  - HBM floor ~110 MB (out 100MB + x strided ~4MB + W2 once) => ~4.7us @ 23.3TB/s.
//  - Dominant modeled cost is W2 re-streaming from L2: (M/MT) * 9.4MB.
//    MT=64 rows/workgroup => 512 WGs * 9.4MB = 4.8GB L2 traffic (4x less than MT=16).
//  - Each B-fragment load now feeds 4 WMMAs (4 m-tiles), each A-fragment 3 WMMAs.
// ---------------------------------------------------------------------------

typedef float v2f __attribute__((ext_vector_type(2)));
typedef float v8f __attribute__((ext_vector_type(8)));

// Problem dims (match reference)
constexpr int BS   = 8 * 4096;   // 32768 token rows (M)
constexpr int EMB  = 768;        // E
constexpr int FFN  = 3072;       // F
constexpr int NQ   = 8;          // quantum features per token

// Tiling
constexpr int MT     = 64;             // rows per workgroup (4 WMMA m-tiles)
constexpr int MTILES = MT / 16;        // 4
constexpr int FC     = 192;            // F-chunk staged in LDS (3072/192 = 16 chunks)
constexpr int FPAD   = FC + 4;         // padded LDS row stride: conflict-free b64 reads
constexpr int BLOCK  = 512;            // 16 wave32s
constexpr int NWAVE  = BLOCK / 32;     // 16
constexpr int ETPW   = (EMB / 16) / NWAVE;  // 3 e-tiles per wave (16*3*16 = 768)

static_assert((MT * FC) % BLOCK == 0, "h-chunk must divide evenly");
static_assert(FFN % FC == 0, "chunking must divide FFN");

__global__ __launch_bounds__(BLOCK, 1)
void ffq_fused(const float* __restrict__ x,
               const float* __restrict__ theta,
               const float* __restrict__ W1,
               const float* __restrict__ W2,
               float* __restrict__ out)
{
    __shared__ __align__(16) float zsh[MT * NQ];        // z tile: 64 x 8   (2 KB)
    __shared__ __align__(16) float hsh[MT * FPAD];      // h chunk: 64 x 192 padded (50 KB)

    const int t    = threadIdx.x;
    const int lane = t & 31;
    const int wave = t >> 5;
    const int m0   = blockIdx.x * MT;

    // ---- Phase 0: z[m][q] = cos(x[m][q]) * cos(theta[q]) -------------------
    {
        const int m = t >> 3;        // BLOCK == MT*NQ == 512: one element per thread
        const int q = t & 7;
        zsh[t] = cosf(x[(size_t)(m0 + m) * EMB + q]) * cosf(theta[q]);
    }
    __syncthreads();

    // ---- Per-lane WMMA fragment geometry -----------------------------------
    // A (16x4 f32): lanes 0-15 -> M=lane, K={0,1}; lanes 16-31 -> M=lane-16, K={2,3}
    // B (4x16 f32): lanes 0-15 -> N=lane, K={0,1}; lanes 16-31 -> N=lane-16, K={2,3}
    // C (16x16 f32): VGPR r, lanes 0-15 -> M=r, N=lane; lanes 16-31 -> M=r+8, N=lane-16
    const int nn = lane & 15;          // N index within e-tile / M row within m-tile
    const int kb = (lane >> 4) << 1;   // 0 or 2: lane-half K base

    const float* hrow = &hsh[nn * FPAD + kb];

    const float* w2base[ETPW];
#pragma unroll
    for (int j = 0; j < ETPW; ++j) {
        const int e = (wave * ETPW + j) * 16 + nn;     // output column this lane covers
        w2base[j] = W2 + (size_t)e * FFN + kb;         // + (fc + k0) walks the K dim
    }

    v8f acc[MTILES][ETPW];
#pragma unroll
    for (int mt = 0; mt < MTILES; ++mt)
#pragma unroll
        for (int j = 0; j < ETPW; ++j)
            acc[mt][j] = (v8f){0.f, 0.f, 0.f, 0.f, 0.f, 0.f, 0.f, 0.f};

    // ---- Main loop over F in chunks of FC ----------------------------------
    for (int fc = 0; fc < FFN; fc += FC) {
        // (a) cooperative h-chunk: h[m][fl] = relu( sum_q z[m][q] * W1[fc+fl][q] )
#pragma unroll
        for (int i = 0; i < (MT * FC) / BLOCK; ++i) {
            const int v  = t + i * BLOCK;
            const int m  = v / FC;
            const int fl = v - m * FC;
            const float4* w1 = reinterpret_cast<const float4*>(W1 + (size_t)(fc + fl) * NQ);
            const float4 wa = w1[0];
            const float4 wb = w1[1];
            const float* zr = &zsh[m * NQ];
            float h = zr[0]*wa.x + zr[1]*wa.y + zr[2]*wa.z + zr[3]*wa.w
                    + zr[4]*wb.x + zr[5]*wb.y + zr[6]*wb.z + zr[7]*wb.w;
            hsh[m * FPAD + fl] = fmaxf(h, 0.0f);
        }
        __syncthreads();

        // (b) warm caches for this wave's W2 streams (global_prefetch_b8)
#pragma unroll
        for (int j = 0; j < ETPW; ++j)
            __builtin_prefetch(w2base[j] + fc, 0, 0);

        // (c) rank-4 WMMA updates over the chunk: 4 A-frags + 3 B-frags -> 12 WMMAs
#pragma unroll 2
        for (int k0 = 0; k0 < FC; k0 += 4) {
            v2f a[MTILES];
#pragma unroll
            for (int mt = 0; mt < MTILES; ++mt)
                a[mt] = *reinterpret_cast<const v2f*>(hrow + mt * 16 * FPAD + k0); // ds_load_b64
#pragma unroll
            for (int j = 0; j < ETPW; ++j) {
                const v2f b = *reinterpret_cast<const v2f*>(w2base[j] + fc + k0); // global_load_b64
#pragma unroll
                for (int mt = 0; mt < MTILES; ++mt) {
                    acc[mt][j] = __builtin_amdgcn_wmma_f32_16x16x4_f32(
                        /*neg_a=*/false, a[mt], /*neg_b=*/false, b,
                        /*c_mod=*/(short)0, acc[mt][j],
                        /*reuse_a=*/false, /*reuse_b=*/false);
                }
            }
        }
        __syncthreads();   // protect hsh before next chunk overwrites it
    }

    // ---- Epilogue: scatter C fragments -------------------------------------
    const int hi8 = (lane >> 4) << 3;   // +0 or +8 row half
#pragma unroll
    for (int mt = 0; mt < MTILES; ++mt) {
        const int mrow = m0 + mt * 16 + hi8;
#pragma unroll
        for (int j = 0; j < ETPW; ++j) {
            const int e0 = (wave * ETPW + j) * 16;
#pragma unroll
            for (int r = 0; r < 8; ++r) {
                out[(size_t)(mrow + r) * EMB + e0 + nn] = acc[mt][j][r];
            }
        }
    }
}

extern "C" void kernel_launch(void* const* d_in, const int* in_sizes, int n_in,
                              void* d_out, int out_size, void* d_ws, size_t ws_size,
                              hipStream_t stream) {
    (void)in_sizes; (void)n_in; (void)d_ws; (void)ws_size; (void)out_size;
    const float* x     = (const float*)d_in[0];
    const float* theta = (const float*)d_in[1];
    const float* W1    = (const float*)d_in[2];
    const float* W2    = (const float*)d_in[3];
    float* out = (float*)d_out;

    dim3 grid(BS / MT);    // 512 workgroups, one 64-row m-strip each
    dim3 block(BLOCK);     // 16 wave32s
    hipLaunchKernelGGL(ffq_fused, grid, block, 0, stream, x, theta, W1, W2, out);
}